// LIFSpike_90958817395414
// MI455X (gfx1250) — compile-verified
//
#include <hip/hip_runtime.h>
#include <hip/hip_bf16.h>

// LIF spiking neuron forward, x: [64,128,32,32, T=8] fp32, scan over last axis.
//   u = TAU*u*(1-o_prev) + x_t ;  o = (u - VTH > 0) ? 1 : 0
// Memory-bound streaming kernel: 537 MB total traffic -> ~23 us floor at
// 23.3 TB/s HBM. One thread per neuron, T=8 contiguous floats = 2x b128
// NT loads + 2x b128 NT stores per thread (zero reuse; don't thrash the
// 192 MB L2 with 512 MB of streamed data).

#define LIF_VTH 1.5f

// Native clang vector type: accepted by __builtin_nontemporal_* (HIP's
// float4 is a struct and is rejected). Same 16B layout -> b128 vmem ops.
typedef float v4f __attribute__((ext_vector_type(4)));

// Un-fusable 0.1f * u: the reference computes rn(rn(TAU*u) + x); letting the
// backend contract mul+add into v_fma perturbs u by 1 ulp, which can flip a
// threshold-adjacent spike and corrupt the rest of that neuron's scan.
// VOP2 v_mul_f32 with the 0.1f literal (0x3dcccccd) in SRC0.
__device__ __forceinline__ float mul_tau_exact(float u) {
    float mu;
    asm("v_mul_f32_e32 %0, 0x3dcccccd, %1" : "=v"(mu) : "v"(u));
    return mu;
}

__global__ __launch_bounds__(256) void lif_fwd_kernel(
    const float* __restrict__ x, float* __restrict__ y, int n_neurons) {
#pragma clang fp contract(off)
    int i = blockIdx.x * blockDim.x + threadIdx.x;
    if (i >= n_neurons) return;

    const v4f* xp = reinterpret_cast<const v4f*>(x) + (size_t)i * 2;
    v4f*       yp = reinterpret_cast<v4f*>(y)       + (size_t)i * 2;

    // Two 128-bit non-temporal loads: 8 timesteps of this neuron.
    v4f a = __builtin_nontemporal_load(xp);
    v4f b = __builtin_nontemporal_load(xp + 1);

    float in[8] = {a.x, a.y, a.z, a.w, b.x, b.y, b.z, b.w};
    float out[8];

    float u = 0.0f;
    float o = 0.0f;
#pragma unroll
    for (int t = 0; t < 8; ++t) {
        // o==1 -> u = x_t exactly (reference: 0.1*u*0 + x).
        // o==0 -> u = rn(rn(0.1*u) + x_t); mul kept opaque so no FMA fusion.
        float decayed = mul_tau_exact(u) + in[t];
        u = (o != 0.0f) ? in[t] : decayed;
        // (u - 1.5f > 0) <=> (u > 1.5f): subtraction near 1.5 is exact
        // (Sterbenz) and rn preserves sign at these magnitudes.
        o = (u > LIF_VTH) ? 1.0f : 0.0f;
        out[t] = o;
    }

    v4f r0 = {out[0], out[1], out[2], out[3]};
    v4f r1 = {out[4], out[5], out[6], out[7]};
    __builtin_nontemporal_store(r0, yp);
    __builtin_nontemporal_store(r1, yp + 1);
}

extern "C" void kernel_launch(void* const* d_in, const int* in_sizes, int n_in,
                              void* d_out, int out_size, void* d_ws, size_t ws_size,
                              hipStream_t stream) {
    const float* x = (const float*)d_in[0];
    float*       y = (float*)d_out;

    const int T = 8;
    int n_neurons = in_sizes[0] / T;  // 64*128*32*32 = 8,388,608

    const int block = 256;            // 8 wave32s per block
    int grid = (n_neurons + block - 1) / block;
    lif_fwd_kernel<<<grid, block, 0, stream>>>(x, y, n_neurons);
}